// DLPCNNLoss_45861660787460
// MI455X (gfx1250) — compile-verified
//
#include <hip/hip_runtime.h>

// Problem constants (match reference)
#define BB    2048
#define DD    2000
#define CC    7
#define KK    20
#define LAMDA 0.003f

typedef __attribute__((ext_vector_type(2))) float v2f;  // f32 A/B frag (16x4 / 4x16)
typedef __attribute__((ext_vector_type(8))) float v8f;  // f32 C/D frag (16x16)

// ---------------------------------------------------------------------------
// Kernel 1: sq[i] = sum_d x[i][d]^2
// ---------------------------------------------------------------------------
__global__ __launch_bounds__(128) void sq_kernel(const float* __restrict__ x,
                                                 float* __restrict__ sq) {
  __shared__ float red[128];
  const int i = blockIdx.x;
  const float* row = x + (size_t)i * DD;
  float s = 0.f;
  for (int d = threadIdx.x; d < DD; d += 128) {
    float v = row[d];
    s += v * v;
  }
  red[threadIdx.x] = s;
  __syncthreads();
  for (int off = 64; off > 0; off >>= 1) {
    if (threadIdx.x < off) red[threadIdx.x] += red[threadIdx.x + off];
    __syncthreads();
  }
  if (threadIdx.x == 0) sq[i] = red[0];
}

// ---------------------------------------------------------------------------
// Kernel 2: Gram matrix via V_WMMA_F32_16X16X4_F32, fused masked-distance
// epilogue:  d2[i][j] = (y[i]==y[j] && i!=j) ? sq[i]+sq[j]-2*G[i][j] : +inf
//
// One wave computes a 32(M) x 64(N) tile: 2 A fragments + 4 B fragments per
// k-step feed 8 WMMAs (16.4 KFLOP per 1.5 KB of wave-level loads, ~10.7
// FLOP/byte from L2 -- up from 6.4 in the 16x64 version).
//
// ISA layouts (CDNA5 ISA, 32-bit 16x4 A; B/C by symmetry):
//   A 16x4 :  lane l -> M = l%16 ; VGPR j -> K = 2*(l/16) + j
//   B 4x16 :  lane l -> N = l%16 ; VGPR j -> K = 2*(l/16) + j
//   C 16x16:  lane l -> N = l%16 ; VGPR v -> M = v + 8*(l/16)
// For G = X * X^T both fragments load as X[row + l%16][k + 2*(l/16) + j],
// i.e. one global b64 load per fragment per k-step (row stride 8000B -> the
// 2-float load is 8B aligned since k is a multiple of 4).
// ---------------------------------------------------------------------------
__global__ __launch_bounds__(256) void gram_kernel(const float* __restrict__ x,
                                                   const float* __restrict__ sq,
                                                   const int* __restrict__ y,
                                                   float* __restrict__ d2) {
  const int lane   = threadIdx.x & 31;
  const int wave   = blockIdx.x * (blockDim.x >> 5) + (threadIdx.x >> 5);
  const int mTile  = wave >> 5;   // 0..63  (2048/32 M-tiles)
  const int nStrip = wave & 31;   // 0..31  (2048/64 N-strips)
  const int m0 = mTile * 32;
  const int n0 = nStrip * 64;
  const int r  = lane & 15;
  const int h  = lane >> 4;

  const float* a0p = x + (size_t)(m0 +  0 + r) * DD + 2 * h;
  const float* a1p = x + (size_t)(m0 + 16 + r) * DD + 2 * h;
  const float* b0p = x + (size_t)(n0 +  0 + r) * DD + 2 * h;
  const float* b1p = x + (size_t)(n0 + 16 + r) * DD + 2 * h;
  const float* b2p = x + (size_t)(n0 + 32 + r) * DD + 2 * h;
  const float* b3p = x + (size_t)(n0 + 48 + r) * DD + 2 * h;

  v8f acc0_0 = {}, acc0_1 = {}, acc0_2 = {}, acc0_3 = {};
  v8f acc1_0 = {}, acc1_1 = {}, acc1_2 = {}, acc1_3 = {};
  for (int k = 0; k < DD; k += 4) {
    v2f a0 = *(const v2f*)(a0p + k);
    v2f a1 = *(const v2f*)(a1p + k);
    v2f b0 = *(const v2f*)(b0p + k);
    v2f b1 = *(const v2f*)(b1p + k);
    v2f b2 = *(const v2f*)(b2p + k);
    v2f b3 = *(const v2f*)(b3p + k);
    acc0_0 = __builtin_amdgcn_wmma_f32_16x16x4_f32(false, a0, false, b0, (short)0, acc0_0, false, false);
    acc1_0 = __builtin_amdgcn_wmma_f32_16x16x4_f32(false, a1, false, b0, (short)0, acc1_0, false, false);
    acc0_1 = __builtin_amdgcn_wmma_f32_16x16x4_f32(false, a0, false, b1, (short)0, acc0_1, false, false);
    acc1_1 = __builtin_amdgcn_wmma_f32_16x16x4_f32(false, a1, false, b1, (short)0, acc1_1, false, false);
    acc0_2 = __builtin_amdgcn_wmma_f32_16x16x4_f32(false, a0, false, b2, (short)0, acc0_2, false, false);
    acc1_2 = __builtin_amdgcn_wmma_f32_16x16x4_f32(false, a1, false, b2, (short)0, acc1_2, false, false);
    acc0_3 = __builtin_amdgcn_wmma_f32_16x16x4_f32(false, a0, false, b3, (short)0, acc0_3, false, false);
    acc1_3 = __builtin_amdgcn_wmma_f32_16x16x4_f32(false, a1, false, b3, (short)0, acc1_3, false, false);
  }

  // Fused masked-distance epilogue (two 16-row M-subtiles)
  float sqi[2][8];
  int   yi[2][8];
#pragma unroll
  for (int s = 0; s < 2; s++) {
    const int mb = m0 + 16 * s + 8 * h;
#pragma unroll
    for (int v = 0; v < 8; v++) {
      sqi[s][v] = sq[mb + v];
      yi[s][v]  = y[mb + v];
    }
  }

  auto emit = [&](int s, int t, const v8f& acc) {
    const int mb  = m0 + 16 * s + 8 * h;  // VGPR v -> row M = mb + v
    const int j   = n0 + t * 16 + r;
    const float sqj = sq[j];
    const int   yj  = y[j];
#pragma unroll
    for (int v = 0; v < 8; v++) {
      const int i = mb + v;
      float dd = sqi[s][v] + sqj - 2.0f * acc[v];
      bool same = (yi[s][v] == yj) && (i != j);
      d2[(size_t)i * BB + j] = same ? dd : __builtin_inff();
    }
  };
  emit(0, 0, acc0_0); emit(0, 1, acc0_1); emit(0, 2, acc0_2); emit(0, 3, acc0_3);
  emit(1, 0, acc1_0); emit(1, 1, acc1_1); emit(1, 2, acc1_2); emit(1, 3, acc1_3);
}

// ---------------------------------------------------------------------------
// Kernel 3: per-row top-K (smallest) indices via insertion into sorted list.
// One thread per row; masked matrix is symmetric + L2-resident so the
// strided access pattern is cheap.
// ---------------------------------------------------------------------------
__global__ __launch_bounds__(256) void topk_kernel(const float* __restrict__ d2,
                                                   int* __restrict__ idx_out) {
  const int i = blockIdx.x * blockDim.x + threadIdx.x;
  if (i >= BB) return;
  float bv[KK];
  int   bi[KK];
#pragma unroll
  for (int t = 0; t < KK; t++) { bv[t] = __builtin_inff(); bi[t] = 0; }
  const float* row = d2 + (size_t)i * BB;
  for (int j = 0; j < BB; j++) {
    float v = row[j];
    if (v < bv[KK - 1]) {
      int p = KK - 1;
      while (p > 0 && bv[p - 1] > v) {
        bv[p] = bv[p - 1];
        bi[p] = bi[p - 1];
        p--;
      }
      bv[p] = v;
      bi[p] = j;
    }
  }
#pragma unroll
  for (int t = 0; t < KK; t++) idx_out[i * KK + t] = bi[t];
}

// ---------------------------------------------------------------------------
// Kernel 4: centers + sum of ||x_i - mean(x[nbrs])||^2 (block per row)
// ---------------------------------------------------------------------------
__global__ __launch_bounds__(256) void center_kernel(const float* __restrict__ x,
                                                     const int* __restrict__ idx,
                                                     float* __restrict__ acc_lp) {
  __shared__ int   nb[KK];
  __shared__ float red[256];
  const int i = blockIdx.x;
  if (threadIdx.x < KK) nb[threadIdx.x] = idx[i * KK + threadIdx.x];
  __syncthreads();

  float local = 0.f;
  const float* xi = x + (size_t)i * DD;
  for (int d = threadIdx.x; d < DD; d += 256) {
    float s = 0.f;
#pragma unroll
    for (int t = 0; t < KK; t++) s += x[(size_t)nb[t] * DD + d];
    float diff = xi[d] - s * (1.0f / (float)KK);
    local += diff * diff;
  }
  red[threadIdx.x] = local;
  __syncthreads();
  for (int off = 128; off > 0; off >>= 1) {
    if (threadIdx.x < off) red[threadIdx.x] += red[threadIdx.x + off];
    __syncthreads();
  }
  if (threadIdx.x == 0) atomicAdd(acc_lp, red[0]);
}

// ---------------------------------------------------------------------------
// Kernel 5: cross-entropy sum (thread per row, wave32 shuffle reduce)
// ---------------------------------------------------------------------------
__global__ __launch_bounds__(256) void ce_kernel(const float* __restrict__ preds,
                                                 const int* __restrict__ y,
                                                 float* __restrict__ acc_ce) {
  const int i = blockIdx.x * blockDim.x + threadIdx.x;
  float val = 0.f;
  if (i < BB) {
    const float* p = preds + (size_t)i * CC;
    float m = p[0];
#pragma unroll
    for (int c = 1; c < CC; c++) m = fmaxf(m, p[c]);
    float s = 0.f;
#pragma unroll
    for (int c = 0; c < CC; c++) s += __expf(p[c] - m);
    float logp = p[y[i]] - m - __logf(s);
    val = -logp;
  }
  // wave32 reduce
  for (int off = 16; off > 0; off >>= 1) val += __shfl_down(val, off, 32);
  if ((threadIdx.x & 31) == 0) atomicAdd(acc_ce, val);
}

// ---------------------------------------------------------------------------
// Kernel 6: final scalar
// ---------------------------------------------------------------------------
__global__ void finalize_kernel(const float* __restrict__ acc,
                                float* __restrict__ out) {
  const float loss_lp = acc[0] / (float)BB;
  const float ce      = acc[1] / (float)BB;
  out[0] = LAMDA * loss_lp * 0.5f + ce;
}

// ---------------------------------------------------------------------------
// Launcher. Inputs: preds [B,C] f32, x [B,D] f32, y [B] i32. Output: scalar.
// Workspace layout (all rewritten every call; accumulators zeroed async):
//   [0, 16MB)      d2 masked distance matrix
//   +BB*BB*4       sq   (BB floats)
//   +BB*4          idx  (BB*KK ints)
//   +BB*KK*4       acc  (2 floats: loss_lp sum, ce sum)
// ---------------------------------------------------------------------------
extern "C" void kernel_launch(void* const* d_in, const int* in_sizes, int n_in,
                              void* d_out, int out_size, void* d_ws, size_t ws_size,
                              hipStream_t stream) {
  const float* preds = (const float*)d_in[0];
  const float* x     = (const float*)d_in[1];
  const int*   y     = (const int*)d_in[2];
  float*       out   = (float*)d_out;

  char* ws = (char*)d_ws;
  float* d2  = (float*)ws;
  float* sq  = (float*)(ws + (size_t)BB * BB * 4);
  int*   idx = (int*)  (ws + (size_t)BB * BB * 4 + (size_t)BB * 4);
  float* acc = (float*)(ws + (size_t)BB * BB * 4 + (size_t)BB * 4 + (size_t)BB * KK * 4);

  hipMemsetAsync(acc, 0, 2 * sizeof(float), stream);

  sq_kernel<<<BB, 128, 0, stream>>>(x, sq);
  // 2048 waves total: 64 M-tiles (32 rows) x 32 N-strips (64 cols);
  // 8 waves per 256-thread block
  gram_kernel<<<256, 256, 0, stream>>>(x, sq, y, d2);
  topk_kernel<<<BB / 256, 256, 0, stream>>>(d2, idx);
  center_kernel<<<BB, 256, 0, stream>>>(x, idx, acc);
  ce_kernel<<<BB / 256, 256, 0, stream>>>(preds, y, acc + 1);
  finalize_kernel<<<1, 1, 0, stream>>>(acc, out);
}